// SSIM_with_patch_mask_5961414607475
// MI455X (gfx1250) — compile-verified
//
#include <hip/hip_runtime.h>
#include <hip/hip_bf16.h>

typedef float v2f __attribute__((ext_vector_type(2)));
typedef float v8f __attribute__((ext_vector_type(8)));

#define B_   16
#define C_   3
#define H_   512
#define W_   512
#define WS_  11
#define OH_  502          // H - WS + 1
#define NT_  32           // ceil(502/16) tiles per dim
#define PB_  32           // 512/16 patches per dim

// workspace layout (float offsets)
#define OFS_CONST 0                      // [0]=C1 [1]=C2
#define OFS_MIN   2                      // 1024
#define OFS_MAX   (2 + 1024)             // 1024
#define OFS_SMALL 2050                   // 16*32*32 = 16384
#define OFS_DIL   (2050 + 16384)         // 16384
#define OFS_PART  (2050 + 2 * 16384)     // B_*C_*NT_*NT_ = 49152
#define NPART     (B_ * C_ * NT_ * NT_)

// ---------------- global min/max of img1 ----------------
__global__ void minmax_part_kernel(const float* __restrict__ x, float* __restrict__ ws) {
    __shared__ float smn[256], smx[256];
    int tid = threadIdx.x;
    long n = (long)B_ * C_ * H_ * W_;
    long i = (long)blockIdx.x * 256 + tid;
    long stride = (long)gridDim.x * 256;
    float mn = 3.4e38f, mx = -3.4e38f;
    for (; i < n; i += stride) { float v = x[i]; mn = fminf(mn, v); mx = fmaxf(mx, v); }
    smn[tid] = mn; smx[tid] = mx; __syncthreads();
    for (int s = 128; s > 0; s >>= 1) {
        if (tid < s) { smn[tid] = fminf(smn[tid], smn[tid + s]); smx[tid] = fmaxf(smx[tid], smx[tid + s]); }
        __syncthreads();
    }
    if (tid == 0) { ws[OFS_MIN + blockIdx.x] = smn[0]; ws[OFS_MAX + blockIdx.x] = smx[0]; }
}

__global__ void minmax_final_kernel(float* __restrict__ ws) {
    __shared__ float smn[256], smx[256];
    int tid = threadIdx.x;
    float mn = 3.4e38f, mx = -3.4e38f;
    for (int i = tid; i < 1024; i += 256) { mn = fminf(mn, ws[OFS_MIN + i]); mx = fmaxf(mx, ws[OFS_MAX + i]); }
    smn[tid] = mn; smx[tid] = mx; __syncthreads();
    for (int s = 128; s > 0; s >>= 1) {
        if (tid < s) { smn[tid] = fminf(smn[tid], smn[tid + s]); smx[tid] = fmaxf(smx[tid], smx[tid + s]); }
        __syncthreads();
    }
    if (tid == 0) {
        float maxv = (smx[0] > 128.0f) ? 255.0f : 1.0f;
        float minv = (smn[0] < -0.5f) ? -1.0f : 0.0f;
        float L = maxv - minv;
        float c1 = 0.01f * L; c1 *= c1;
        float c2 = 0.03f * L; c2 *= c2;
        ws[OFS_CONST + 0] = c1;
        ws[OFS_CONST + 1] = c2;
    }
}

// ---------------- patch max-pool of mask ----------------
__global__ void patch_max_kernel(const int* __restrict__ mask, float* __restrict__ ws) {
    int p = blockIdx.x * 256 + threadIdx.x;
    if (p >= B_ * PB_ * PB_) return;
    int b = p >> 10, py = (p >> 5) & 31, px = p & 31;
    const int* base = mask + ((long)b * H_ + py * 16) * W_ + px * 16;
    int any = 0;
    for (int r = 0; r < 16; ++r)
        for (int c = 0; c < 16; ++c) any |= base[r * W_ + c];
    ws[OFS_SMALL + p] = any ? 1.0f : 0.0f;
}

// ---------------- 3x3 binary dilation ----------------
__global__ void dilate_kernel(float* __restrict__ ws) {
    int p = blockIdx.x * 256 + threadIdx.x;
    if (p >= B_ * PB_ * PB_) return;
    int b = p >> 10, py = (p >> 5) & 31, px = p & 31;
    float s = 0.0f;
    for (int dy = -1; dy <= 1; ++dy)
        for (int dx = -1; dx <= 1; ++dx) {
            int y = py + dy, x = px + dx;
            if (y >= 0 && y < 32 && x >= 0 && x < 32)
                s += ws[OFS_SMALL + (b << 10) + (y << 5) + x];
        }
    ws[OFS_DIL + p] = (s > 0.0f) ? 1.0f : 0.0f;
}

// ---------------- main SSIM tile kernel (1 wave / 16x16 output tile) ----------------
__device__ inline v8f wmma4(v2f a, v2f b, v8f c) {
    return __builtin_amdgcn_wmma_f32_16x16x4_f32(false, a, false, b, (short)0, c, false, false);
}

__global__ void __launch_bounds__(32) ssim_tile_kernel(const float* __restrict__ img1,
                                                       const float* __restrict__ img2,
                                                       float* __restrict__ ws) {
    __shared__ float sG[12];                 // 1-D gaussian
    __shared__ float sAPt[14][16][2];        // banded A fragments: sAPt[j][m] = {g[2j-m], g[2j+1-m]} (0 outside band)
    __shared__ float sA[26][28];
    __shared__ float sB[26][28];
    __shared__ float sHp[5][14][16][2];      // pair-interleaved H: (row,n) -> [row>>1][n][row&1]; pair 13 = zero pad

    int lane = threadIdx.x;
    int m = lane & 15;
    int half = lane >> 4;

    // Gaussian window (match numpy: float64 exp, normalize, cast f32)
    if (lane == 0) {
        double g[11], sum = 0.0;
        for (int i = 0; i < 11; ++i) { double xd = (double)(i - 5); g[i] = exp(-(xd * xd) / 4.5); sum += g[i]; }
        for (int i = 0; i < 11; ++i) sG[i] = (float)(g[i] / sum);
        sG[11] = 0.0f;
    }
    __syncthreads();

    // banded A-fragment table (unconditional b64 loads later): 14*16*2 = 448 entries
    for (int idx = lane; idx < 448; idx += 32) {
        int j = idx >> 5, mm = (idx >> 1) & 15, s = idx & 1;
        int d = 2 * j + s - mm;
        sAPt[j][mm][s] = (d >= 0 && d <= 10) ? sG[d] : 0.0f;
    }
    // zero pad pair 13 of sHp (rows 26,27): 5*16*2 = 160 entries
    for (int idx = lane; idx < 160; idx += 32) {
        int q = idx >> 5, mm = (idx >> 1) & 15, s = idx & 1;
        sHp[q][13][mm][s] = 0.0f;
    }

    int blk = blockIdx.x;
    int tx = blk & 31, ty = (blk >> 5) & 31;
    int bc = blk >> 10;            // 0..47
    int cc = bc % 3, b = bc / 3;
    int y0 = ty * 16, x0 = tx * 16;
    const float* pa = img1 + (long)(b * C_ + cc) * H_ * W_;
    const float* pb = img2 + (long)(b * C_ + cc) * H_ * W_;

    // load 26x26 input region (clamped at edges; clamped rows only feed masked-off outputs)
    for (int idx = lane; idx < 26 * 26; idx += 32) {
        int r = idx / 26, c = idx - r * 26;
        int gy = y0 + r; if (gy > H_ - 1) gy = H_ - 1;
        int gx = x0 + c; if (gx > W_ - 1) gx = W_ - 1;
        sA[r][c] = pa[gy * W_ + gx];
        sB[r][c] = pb[gy * W_ + gx];
    }
    __syncthreads();

    // horizontal 11-tap pass, 5 quantities, 26 rows x 16 output cols
    for (int idx = lane; idx < 26 * 16; idx += 32) {
        int r = idx >> 4, x = idx & 15;
        float ha = 0.f, hb = 0.f, haa = 0.f, hbb = 0.f, hab = 0.f;
#pragma unroll
        for (int i = 0; i < 11; ++i) {
            float w = sG[i];
            float va = sA[r][x + i], vb = sB[r][x + i];
            ha += w * va;        hb += w * vb;
            haa += w * va * va;  hbb += w * vb * vb;  hab += w * va * vb;
        }
        int rp = r >> 1, rs = r & 1;
        sHp[0][rp][x][rs] = ha;  sHp[1][rp][x][rs] = hb;  sHp[2][rp][x][rs] = haa;
        sHp[3][rp][x][rs] = hbb; sHp[4][rp][x][rs] = hab;
    }
    __syncthreads();

    // vertical 11-tap pass as banded matmul: Out(16x16) = Gt(16x28) * Hpad(28x16)
    // A 16x4 f32 layout: lanes 0-15 hold M=lane, K={4kc,4kc+1}; lanes 16-31 hold M=lane-16, K={4kc+2,4kc+3}
    // B 4x16 f32 layout (K split across lane halves), C/D per ISA 7.12.2 table.
    v8f acc0 = {}, acc1 = {}, acc2 = {}, acc3 = {}, acc4 = {};
#pragma unroll
    for (int kc = 0; kc < 7; ++kc) {
        int j = 2 * kc + half;                       // row-pair index: rows J0 = 2j, 2j+1
        v2f A  = *(const v2f*)&sAPt[j][m][0];        // aligned ds_load_b64, unconditional
        v2f B0 = *(const v2f*)&sHp[0][j][m][0];
        v2f B1 = *(const v2f*)&sHp[1][j][m][0];
        v2f B2 = *(const v2f*)&sHp[2][j][m][0];
        v2f B3 = *(const v2f*)&sHp[3][j][m][0];
        v2f B4 = *(const v2f*)&sHp[4][j][m][0];
        acc0 = wmma4(A, B0, acc0);
        acc1 = wmma4(A, B1, acc1);
        acc2 = wmma4(A, B2, acc2);
        acc3 = wmma4(A, B3, acc3);
        acc4 = wmma4(A, B4, acc4);
    }

    // SSIM + validity mask + partial sum
    float C1v = ws[OFS_CONST + 0];
    float C2v = ws[OFS_CONST + 1];
    const float* dil = ws + OFS_DIL + (b << 10);
    float partial = 0.0f;
#pragma unroll
    for (int r = 0; r < 8; ++r) {
        int my = r + half * 8;
        int gy = y0 + my, gx = x0 + m;
        if (gy < OH_ && gx < OH_) {
            float mu1 = acc0[r], mu2 = acc1[r];
            float e11 = acc2[r], e22 = acc3[r], e12 = acc4[r];
            float mu1s = mu1 * mu1, mu2s = mu2 * mu2, mu12 = mu1 * mu2;
            float s1 = e11 - mu1s, s2 = e22 - mu2s, s12 = e12 - mu12;
            float v1 = 2.0f * s12 + C2v, v2 = s1 + s2 + C2v;
            float ssim = ((2.0f * mu12 + C1v) * v1) / ((mu1s + mu2s + C1v) * v2);
            int by0 = gy >> 4, by1 = (gy + 10) >> 4;
            int bx0 = gx >> 4, bx1 = (gx + 10) >> 4;
            bool ok = dil[(by0 << 5) + bx0] > 0.f && dil[(by0 << 5) + bx1] > 0.f &&
                      dil[(by1 << 5) + bx0] > 0.f && dil[(by1 << 5) + bx1] > 0.f;
            if (ok) partial += ssim;
        }
    }
    // wave32 butterfly reduction (fixed structure -> deterministic)
#pragma unroll
    for (int off = 16; off > 0; off >>= 1) partial += __shfl_xor(partial, off, 32);
    if (lane == 0) ws[OFS_PART + blk] = partial;
}

// ---------------- final fixed-order reduction ----------------
__global__ void final_reduce_kernel(const float* __restrict__ ws, float* __restrict__ out) {
    __shared__ float sm[256];
    int tid = threadIdx.x;
    float s = 0.0f;
    for (int i = tid; i < NPART; i += 256) s += ws[OFS_PART + i];
    sm[tid] = s; __syncthreads();
    for (int k = 128; k > 0; k >>= 1) {
        if (tid < k) sm[tid] += sm[tid + k];
        __syncthreads();
    }
    if (tid == 0) out[0] = sm[0] * (1.0f / ((float)B_ * C_ * OH_ * OH_));
}

extern "C" void kernel_launch(void* const* d_in, const int* in_sizes, int n_in,
                              void* d_out, int out_size, void* d_ws, size_t ws_size,
                              hipStream_t stream) {
    (void)in_sizes; (void)n_in; (void)out_size; (void)ws_size;
    const float* img1 = (const float*)d_in[0];
    const float* img2 = (const float*)d_in[1];
    const int*   mask = (const int*)d_in[2];
    float* out = (float*)d_out;
    float* ws  = (float*)d_ws;

    minmax_part_kernel<<<1024, 256, 0, stream>>>(img1, ws);
    minmax_final_kernel<<<1, 256, 0, stream>>>(ws);
    patch_max_kernel<<<(B_ * PB_ * PB_ + 255) / 256, 256, 0, stream>>>(mask, ws);
    dilate_kernel<<<(B_ * PB_ * PB_ + 255) / 256, 256, 0, stream>>>(ws);
    ssim_tile_kernel<<<NPART, 32, 0, stream>>>(img1, img2, ws);
    final_reduce_kernel<<<1, 256, 0, stream>>>(ws, out);
}